// FaceWeightedFusion_90159953478115
// MI455X (gfx1250) — compile-verified
//
#include <hip/hip_runtime.h>
#include <cstdint>

#define BB 4
#define HH 1024
#define WW 1024
#define NN 262144
#define RAD 45     // 15 erosion iterations * 3 px radius each
#define DINF 46    // "no zero within RAD"

// mask_erode value as a function of capped Chebyshev distance to nearest zero.
// d==0  -> pixel itself is 0 -> weight 0
// d in (3i, 3(i+1)] -> eroded at iteration i -> rate_i = 0.1 + i*(0.9/15)
// d>45  -> never eroded -> 1.0
__device__ __forceinline__ float weight_from_d(int d) {
    if (d <= 0) return 0.0f;
    if (d > RAD) return 1.0f;
    int i = (d - 1) / 3;
    return (float)(0.1 + (double)i * ((1.0 - 0.1) / 15.0));
}

// Kernel 1: per-row nearest-zero horizontal distance (capped at DINF).
// Row tile + halo staged into LDS via CDNA5 async global->LDS loads
// (ASYNCcnt-tracked), then a per-thread early-exit scan.
__global__ __launch_bounds__(256)
void hdist_kernel(const float* __restrict__ mask, unsigned char* __restrict__ dh) {
    __shared__ float s[RAD + 256 + RAD + 2];   // 346 used, padded
    const int tid = threadIdx.x;
    const int row = blockIdx.y;                // b*H + y (rows independent)
    const int x0  = blockIdx.x * 256;
    const float* rowp = mask + (size_t)row * WW;

    for (int i = tid; i < 256 + 2 * RAD; i += 256) {
        int gx = x0 - RAD + i;
        if (gx >= 0 && gx < WW) {
            uint32_t ldsa = (uint32_t)(uintptr_t)&s[i];  // flat->LDS: low 32 bits
            const float* gp = rowp + gx;
            asm volatile("global_load_async_to_lds_b32 %0, %1, off"
                         :: "v"(ldsa), "v"(gp) : "memory");
        } else {
            s[i] = 1.0f;   // +inf padding of min-pool: border is never a zero
        }
    }
    asm volatile("s_wait_asynccnt 0" ::: "memory");
    __syncthreads();

    int x = x0 + tid;
    if (x >= WW) return;
    int li = tid + RAD;
    int d;
    if (s[li] == 0.0f) {
        d = 0;
    } else {
        d = DINF;
        for (int r = 1; r <= RAD; ++r) {         // E[d] ~ 1 for random mask
            if (s[li - r] == 0.0f || s[li + r] == 0.0f) { d = r; break; }
        }
    }
    dh[(size_t)row * WW + x] = (unsigned char)d;
}

// Kernel 2: vertical combine  d(p) = min_dy max(|dy|, hdist(x, y+dy)), early exit.
__global__ __launch_bounds__(256)
void vdist_kernel(const unsigned char* __restrict__ dh, unsigned char* __restrict__ df) {
    int idx = blockIdx.x * 256 + threadIdx.x;
    if (idx >= BB * HH * WW) return;
    int y = (idx >> 10) & (HH - 1);
    int best = dh[idx];
    for (int r = 1; r <= RAD; ++r) {
        if (best <= r) break;                    // candidates at radius r are >= r
        if (y - r >= 0) {
            int f = dh[idx - r * WW];
            int c = f > r ? f : r;
            if (c < best) best = c;
        }
        if (y + r < HH) {
            int f = dh[idx + r * WW];
            int c = f > r ? f : r;
            if (c < best) best = c;
        }
    }
    df[idx] = (unsigned char)best;
}

// Kernel 3: bilinear gathers (L2-resident textures) + fusion math.
__global__ __launch_bounds__(256)
void fuse_kernel(const float* __restrict__ face_depth,
                 const float* __restrict__ edge_mask,
                 const unsigned char* __restrict__ df,
                 const float* __restrict__ xy,
                 const float* __restrict__ z,
                 const float* __restrict__ occ_body,
                 const float* __restrict__ occ_face,
                 float* __restrict__ out) {
    int i = blockIdx.x * 256 + threadIdx.x;
    if (i >= BB * NN) return;
    int b = i >> 18;           // i / N
    int n = i & (NN - 1);

    float xf = xy[(size_t)b * 2 * NN + n];
    float yf = xy[(size_t)b * 2 * NN + NN + n];
    float x = (xf + 1.0f) * 0.5f * (float)(WW - 1);
    float y = (yf + 1.0f) * 0.5f * (float)(HH - 1);
    float x0f = floorf(x), y0f = floorf(y);
    float wx = x - x0f, wy = y - y0f;
    int x0 = (int)x0f, y0 = (int)y0f;
    int x0i = min(max(x0, 0), WW - 1);
    int x1i = min(max(x0 + 1, 0), WW - 1);
    int y0i = min(max(y0, 0), HH - 1);
    int y1i = min(max(y0 + 1, 0), HH - 1);

    size_t base = (size_t)b * HH * WW;
    size_t i00 = base + (size_t)y0i * WW + x0i;
    size_t i10 = base + (size_t)y0i * WW + x1i;
    size_t i01 = base + (size_t)y1i * WW + x0i;
    size_t i11 = base + (size_t)y1i * WW + x1i;

    float w00 = (1.0f - wx) * (1.0f - wy), w10 = wx * (1.0f - wy);
    float w01 = (1.0f - wx) * wy,          w11 = wx * wy;

    float edge = edge_mask[i00] * w00 + edge_mask[i10] * w10 +
                 edge_mask[i01] * w01 + edge_mask[i11] * w11;
    float eb = (edge > 0.01f) ? 1.0f : 0.0f;

    float depth = (face_depth[i00] * w00 + face_depth[i10] * w10 +
                   face_depth[i01] * w01 + face_depth[i11] * w11) * eb;

    float lab = (weight_from_d(df[i00]) * w00 + weight_from_d(df[i10]) * w10 +
                 weight_from_d(df[i01]) * w01 + weight_from_d(df[i11]) * w11) * eb;

    float psdf = z[i] - depth;
    float w = expf(-(psdf * psdf) * 1000.0f) * lab;
    out[i] = w * occ_face[i] + (1.0f - w) * occ_body[i];
}

extern "C" void kernel_launch(void* const* d_in, const int* in_sizes, int n_in,
                              void* d_out, int out_size, void* d_ws, size_t ws_size,
                              hipStream_t stream) {
    (void)in_sizes; (void)n_in; (void)out_size; (void)ws_size;
    const float* mask     = (const float*)d_in[0];
    const float* depth    = (const float*)d_in[1];
    const float* edge     = (const float*)d_in[2];
    const float* xy       = (const float*)d_in[3];
    const float* z        = (const float*)d_in[4];
    const float* occ_body = (const float*)d_in[5];
    const float* occ_face = (const float*)d_in[6];
    float* out = (float*)d_out;

    unsigned char* dh = (unsigned char*)d_ws;                 // 4 MB horizontal dist
    unsigned char* df = dh + (size_t)BB * HH * WW;            // 4 MB final dist

    dim3 g1(WW / 256, BB * HH);
    hipLaunchKernelGGL(hdist_kernel, g1, dim3(256), 0, stream, mask, dh);

    int npix = BB * HH * WW;
    hipLaunchKernelGGL(vdist_kernel, dim3(npix / 256), dim3(256), 0, stream, dh, df);

    int npts = BB * NN;
    hipLaunchKernelGGL(fuse_kernel, dim3(npts / 256), dim3(256), 0, stream,
                       depth, edge, df, xy, z, occ_body, occ_face, out);
}